// deform_block_feat_middle_22797686407589
// MI455X (gfx1250) — compile-verified
//
#include <hip/hip_runtime.h>

typedef __attribute__((ext_vector_type(2))) float v2f;
typedef __attribute__((ext_vector_type(8))) float v8f;

#define B_    8
#define N_    4096
#define FIN   64
#define FOUT  128
#define KNN   20
#define NPTS  (B_ * N_)        // 32768
#define NEDGE (NPTS * KNN)     // 655360

// stats region: 9 layers x 512 floats: [sum(128) | sumsq(128) | mean(128) | rstd(128)]
// layers: 0=cf1 1=cf2 2=cf3 3=ic 4=c2 5=uc 8=ac
#define STATS_F 4608

__device__ __forceinline__ float lrelu_f(float v) { return v >= 0.f ? v : 0.01f * v; }

__device__ __forceinline__ v8f wmma4(v2f a, v2f b, v8f c) {
  // D = A(16x4, f32) * B(4x16, f32) + C(16x16, f32)
  return __builtin_amdgcn_wmma_f32_16x16x4_f32(false, a, false, b, (short)0, c, false, false);
}

// ---------------- utility kernels ----------------

__global__ void zero_kernel(float* p, int n) {
  const int i = blockIdx.x * 256 + threadIdx.x;
  if (i < n) p[i] = 0.f;
}

__global__ void finalize_kernel(float* stats, int layer, int nch, float count) {
  const int t = threadIdx.x;
  if (t >= nch) return;
  float* base = stats + layer * 512;
  const float mean = base[t] / count;
  const float var  = base[128 + t] / count - mean * mean;
  base[256 + t] = mean;
  base[384 + t] = rsqrtf(var + 1e-5f);
}

// xt[b][n][f] = x[b][f][n]; sq[b][n] = sum_f x^2
__global__ void prep_kernel(const float* __restrict__ x, float* __restrict__ xt,
                            float* __restrict__ sq) {
  const int p = blockIdx.x * 256 + threadIdx.x;
  if (p >= NPTS) return;
  const int b = p >> 12, n = p & (N_ - 1);
  const float* xb = x + (size_t)b * FIN * N_ + n;
  float* row = xt + (size_t)p * FIN;
  float s = 0.f;
  for (int f = 0; f < FIN; ++f) {
    const float v = xb[(size_t)f * N_];
    row[f] = v;
    s += v * v;
  }
  sq[p] = s;
}

// ---------------- fused KNN: WMMA distance tiles + per-row top-20 ----------------
// one wave per 16 query rows; streams 16-col tiles over all 4096 points.
__global__ __launch_bounds__(32)
void knn_kernel(const float* __restrict__ x, const float* __restrict__ xt,
                const float* __restrict__ sq, int* __restrict__ nbr) {
  __shared__ float dt[16][17];
  __shared__ float bd[16][KNN];
  __shared__ int   bi[16][KNN];
  const int lane = threadIdx.x, n16 = lane & 15, kh = lane >> 4;
  const int b  = blockIdx.x >> 8;
  const int i0 = (blockIdx.x & 255) << 4;
  const float* xb  = x  + (size_t)b * FIN * N_;
  const float* sqb = sq + ((size_t)b << 12);

  for (int t = lane; t < 16 * KNN; t += 32) { bd[t / KNN][t % KNN] = 3.4e38f; bi[t / KNN][t % KNN] = 0; }

  // preload A fragments for rows i0..i0+15 over all 64 features (16 k-steps of 4)
  v2f afr[16];
  const float* arow = xt + (((size_t)b << 12) + i0 + n16) * FIN;
#pragma unroll
  for (int t = 0; t < 16; ++t) { afr[t].x = arow[4 * t + 2 * kh]; afr[t].y = arow[4 * t + 2 * kh + 1]; }
  float sqi[8];
#pragma unroll
  for (int r = 0; r < 8; ++r) sqi[r] = sqb[i0 + r + 8 * kh];
  __syncthreads();

  for (int j0 = 0; j0 < N_; j0 += 16) {
    v8f acc = {};
#pragma unroll
    for (int t = 0; t < 16; ++t) {
      v2f bb;  // B[k][j] = x[b][k][j0+j]  (feature-major rows are contiguous in x)
      bb.x = xb[(size_t)(4 * t + 2 * kh) * N_ + j0 + n16];
      bb.y = xb[(size_t)(4 * t + 2 * kh + 1) * N_ + j0 + n16];
      acc = wmma4(afr[t], bb, acc);
    }
    const float sqj = sqb[j0 + n16];
#pragma unroll
    for (int r = 0; r < 8; ++r) {
      const int m = r + 8 * kh;
      float d = sqi[r] + sqj - 2.f * acc[r];
      if (i0 + m == j0 + n16) d = 3.4e38f;  // exclude self
      dt[m][n16] = d;
    }
    __syncthreads();
    if (lane < 16) {
#pragma unroll 1
      for (int jj = 0; jj < 16; ++jj) {
        const float d = dt[lane][jj];
        if (d < bd[lane][KNN - 1]) {
          int pos = KNN - 1;
          while (pos > 0 && bd[lane][pos - 1] > d) {
            bd[lane][pos] = bd[lane][pos - 1]; bi[lane][pos] = bi[lane][pos - 1]; --pos;
          }
          bd[lane][pos] = d; bi[lane][pos] = j0 + jj;
        }
      }
    }
    __syncthreads();
  }
  if (lane < 16)
    for (int kk = 0; kk < KNN; ++kk)
      nbr[(((size_t)b << 12) + i0 + lane) * KNN + kk] = bi[lane][kk];
}

// ---------------- per-edge MLP chain, multi-pass for BN stats ----------------
// STAGE 1: stats(cf1).  STAGE 2: stats(cf2).  STAGE 3: stats(cf3)+stats(ic).
// STAGE 4: softmax attention, inte, c2 contraction -> y + stats(c2).
// One wave per 8 points, taken in turn (20 edges + 12 zero pad rows -> two 16-row M tiles).
template <int STAGE>
__global__ __launch_bounds__(32)
void edge_stage_kernel(const float* __restrict__ xt, const int* __restrict__ nbr,
                       const float* __restrict__ W1, const float* __restrict__ B1c,
                       const float* __restrict__ G1, const float* __restrict__ BE1,
                       const float* __restrict__ W2, const float* __restrict__ B2c,
                       const float* __restrict__ G2, const float* __restrict__ BE2,
                       const float* __restrict__ W3, const float* __restrict__ B3c,
                       const float* __restrict__ G3, const float* __restrict__ BE3,
                       const float* __restrict__ WI, const float* __restrict__ BIc,
                       const float* __restrict__ GI, const float* __restrict__ BEI,
                       const float* __restrict__ C2W, const float* __restrict__ C2B,
                       float* __restrict__ stats, float* __restrict__ y) {
  const int lane = threadIdx.x;
  const int n16 = lane & 15;
  const int kh  = lane >> 4;

  __shared__ float E[32][132];
  __shared__ float Cc[64];
  __shared__ float H1[32][17];
  __shared__ float H2[32][65];
  __shared__ float A3[32][65];
  __shared__ float AIC[32][65];

  // zero pad rows 20..31 of E once (never rewritten)
  for (int t = lane; t < 12 * 128; t += 32) E[20 + (t >> 7)][t & 127] = 0.f;

  // hoisted BN params for already-finalized layers (channels fixed per lane)
  float m1v = 0.f, r1v = 0.f, g1v = 0.f, be1v = 0.f;
  if (STAGE >= 2) {
    m1v = stats[0 * 512 + 256 + n16]; r1v = stats[0 * 512 + 384 + n16];
    g1v = G1[n16]; be1v = BE1[n16];
  }
  float m2v[4] = {0, 0, 0, 0}, r2v[4] = {0, 0, 0, 0}, g2v[4] = {0, 0, 0, 0}, be2v[4] = {0, 0, 0, 0};
  if (STAGE >= 3)
    for (int nt = 0; nt < 4; ++nt) {
      const int ch = nt * 16 + n16;
      m2v[nt] = stats[1 * 512 + 256 + ch]; r2v[nt] = stats[1 * 512 + 384 + ch];
      g2v[nt] = G2[ch]; be2v[nt] = BE2[ch];
    }
  float m3v[4] = {0, 0, 0, 0}, r3v[4] = {0, 0, 0, 0}, g3v[4] = {0, 0, 0, 0}, be3v[4] = {0, 0, 0, 0};
  if (STAGE >= 4)
    for (int nt = 0; nt < 4; ++nt) {
      const int ch = nt * 16 + n16;
      m3v[nt] = stats[2 * 512 + 256 + ch]; r3v[nt] = stats[2 * 512 + 384 + ch];
      g3v[nt] = G3[ch]; be3v[nt] = BE3[ch];
    }

  float s1 = 0.f, q1 = 0.f;
  float s2[4] = {0, 0, 0, 0}, q2[4] = {0, 0, 0, 0};
  float s3[4] = {0, 0, 0, 0}, q3[4] = {0, 0, 0, 0};
  float si[4] = {0, 0, 0, 0}, qi[4] = {0, 0, 0, 0};
  float sy[4] = {0, 0, 0, 0}, qy[4] = {0, 0, 0, 0};

  for (int pi = 0; pi < 8; ++pi) {
    const int p = blockIdx.x * 8 + pi;
    const int b = p >> 12;
    __syncthreads();
    const float* cen = xt + (size_t)p * FIN;
    for (int c = lane; c < FIN; c += 32) Cc[c] = cen[c];
    __syncthreads();
    // edge features: ee = [central, neighbor - central]
    for (int k = 0; k < KNN; ++k) {
      const int j = nbr[(size_t)p * KNN + k];  // uniform per wave
      const float* nb = xt + (((size_t)b << 12) + j) * FIN;
      for (int c = lane; c < 128; c += 32)
        E[k][c] = (c < 64) ? Cc[c] : (nb[c - 64] - Cc[c - 64]);
    }
    __syncthreads();

    // ---- layer 1: 128 -> 16 ----
    for (int m0 = 0; m0 < 32; m0 += 16) {
      v8f acc = {};
      for (int k0 = 0; k0 < 128; k0 += 4) {
        v2f a, bb;
        a.x  = E[m0 + n16][k0 + 2 * kh];      a.y  = E[m0 + n16][k0 + 2 * kh + 1];
        bb.x = W1[n16 * 128 + k0 + 2 * kh];   bb.y = W1[n16 * 128 + k0 + 2 * kh + 1];
        acc = wmma4(a, bb, acc);
      }
#pragma unroll
      for (int r = 0; r < 8; ++r) {
        const int m = m0 + r + 8 * kh;
        const float z = acc[r] + B1c[n16];
        if (STAGE == 1) { if (m < KNN) { s1 += z; q1 += z * z; } }
        else H1[m][n16] = lrelu_f((z - m1v) * r1v * g1v + be1v);
      }
    }

    if (STAGE >= 2) {
      __syncthreads();
      // ---- layer 2: 16 -> 64 ----
      for (int m0 = 0; m0 < 32; m0 += 16)
        for (int nt = 0; nt < 4; ++nt) {
          v8f acc = {};
          const int ch = nt * 16 + n16;
          for (int k0 = 0; k0 < 16; k0 += 4) {
            v2f a, bb;
            a.x  = H1[m0 + n16][k0 + 2 * kh];   a.y  = H1[m0 + n16][k0 + 2 * kh + 1];
            bb.x = W2[ch * 16 + k0 + 2 * kh];   bb.y = W2[ch * 16 + k0 + 2 * kh + 1];
            acc = wmma4(a, bb, acc);
          }
#pragma unroll
          for (int r = 0; r < 8; ++r) {
            const int m = m0 + r + 8 * kh;
            const float z = acc[r] + B2c[ch];
            if (STAGE == 2) { if (m < KNN) { s2[nt] += z; q2[nt] += z * z; } }
            else H2[m][ch] = lrelu_f((z - m2v[nt]) * r2v[nt] * g2v[nt] + be2v[nt]);
          }
        }
    }

    if (STAGE >= 3) {
      __syncthreads();
      // ---- layer 3 (cf_w3): 64 -> 64 ----
      for (int m0 = 0; m0 < 32; m0 += 16)
        for (int nt = 0; nt < 4; ++nt) {
          v8f acc = {};
          const int ch = nt * 16 + n16;
          for (int k0 = 0; k0 < 64; k0 += 4) {
            v2f a, bb;
            a.x  = H2[m0 + n16][k0 + 2 * kh];   a.y  = H2[m0 + n16][k0 + 2 * kh + 1];
            bb.x = W3[ch * 64 + k0 + 2 * kh];   bb.y = W3[ch * 64 + k0 + 2 * kh + 1];
            acc = wmma4(a, bb, acc);
          }
#pragma unroll
          for (int r = 0; r < 8; ++r) {
            const int m = m0 + r + 8 * kh;
            const float z = acc[r] + B3c[ch];
            if (STAGE == 3) { if (m < KNN) { s3[nt] += z; q3[nt] += z * z; } }
            else A3[m][ch] = lrelu_f((z - m3v[nt]) * r3v[nt] * g3v[nt] + be3v[nt]);
          }
        }
      // ---- interaction conv: 128 -> 64 (from E) ----
      for (int m0 = 0; m0 < 32; m0 += 16)
        for (int nt = 0; nt < 4; ++nt) {
          v8f acc = {};
          const int ch = nt * 16 + n16;
          for (int k0 = 0; k0 < 128; k0 += 4) {
            v2f a, bb;
            a.x  = E[m0 + n16][k0 + 2 * kh];     a.y  = E[m0 + n16][k0 + 2 * kh + 1];
            bb.x = WI[ch * 128 + k0 + 2 * kh];   bb.y = WI[ch * 128 + k0 + 2 * kh + 1];
            acc = wmma4(a, bb, acc);
          }
#pragma unroll
          for (int r = 0; r < 8; ++r) {
            const int m = m0 + r + 8 * kh;
            const float z = acc[r] + BIc[ch];
            if (STAGE == 3) { if (m < KNN) { si[nt] += z; qi[nt] += z * z; } }
            else AIC[m][ch] = z;  // raw; BN applied in attention step
          }
        }
    }

    if (STAGE == 4) {
      __syncthreads();
      // softmax over k per channel; inte = lrelu(bn_ic(z_ic)) * w_att -> AIC
#pragma unroll
      for (int cc = 0; cc < 2; ++cc) {
        const int c = lane + cc * 32;
        const float mi = stats[3 * 512 + 256 + c], ri = stats[3 * 512 + 384 + c];
        const float gi = GI[c], bei = BEI[c];
        float mx = -3.4e38f;
#pragma unroll
        for (int k = 0; k < KNN; ++k) mx = fmaxf(mx, A3[k][c]);
        float ex[KNN];
        float ssum = 0.f;
#pragma unroll
        for (int k = 0; k < KNN; ++k) { ex[k] = __expf(A3[k][c] - mx); ssum += ex[k]; }
        const float inv = 1.f / ssum;
#pragma unroll
        for (int k = 0; k < KNN; ++k) {
          const float w  = ex[k] * inv;
          const float zi = AIC[k][c];
          AIC[k][c] = lrelu_f((zi - mi) * ri * gi + bei) * w;
        }
      }
      __syncthreads();
      // y[o] = sum_{k,c} inte[k][c] * c2_w[o,c,k] + c2_b[o]
#pragma unroll
      for (int oo = 0; oo < 4; ++oo) {
        const int o = lane + oo * 32;
        float acc = C2B[o];
        for (int c = 0; c < 64; ++c) {
          const float* wrow = C2W + ((size_t)o * 64 + c) * KNN;
#pragma unroll
          for (int k = 0; k < KNN; ++k) acc += AIC[k][c] * wrow[k];
        }
        y[(size_t)p * FOUT + o] = acc;
        sy[oo] += acc; qy[oo] += acc * acc;
      }
    }
  }
  __syncthreads();

  if (STAGE == 1) {
    atomicAdd(stats + 0 * 512 + n16, s1);
    atomicAdd(stats + 0 * 512 + 128 + n16, q1);
  }
  if (STAGE == 2)
    for (int nt = 0; nt < 4; ++nt) {
      const int ch = nt * 16 + n16;
      atomicAdd(stats + 1 * 512 + ch, s2[nt]);
      atomicAdd(stats + 1 * 512 + 128 + ch, q2[nt]);
    }
  if (STAGE == 3)
    for (int nt = 0; nt < 4; ++nt) {
      const int ch = nt * 16 + n16;
      atomicAdd(stats + 2 * 512 + ch, s3[nt]);
      atomicAdd(stats + 2 * 512 + 128 + ch, q3[nt]);
      atomicAdd(stats + 3 * 512 + ch, si[nt]);
      atomicAdd(stats + 3 * 512 + 128 + ch, qi[nt]);
    }
  if (STAGE == 4)
    for (int oo = 0; oo < 4; ++oo) {
      const int o = lane + oo * 32;
      atomicAdd(stats + 4 * 512 + o, sy[oo]);
      atomicAdd(stats + 4 * 512 + 128 + o, qy[oo]);
    }
}

// ---------------- y2 = relu(bn_c2(y)) + uc stats ----------------
__global__ __launch_bounds__(128)
void y2_kernel(const float* __restrict__ y, const float* __restrict__ c2g,
               const float* __restrict__ c2be, float* __restrict__ stats,
               float* __restrict__ y2) {
  const int ch = threadIdx.x;  // blockDim == 128 -> ch == i % 128
  const float m = stats[4 * 512 + 256 + ch], r = stats[4 * 512 + 384 + ch];
  const float g = c2g[ch], be = c2be[ch];
  float s = 0.f, q = 0.f;
  const size_t stride = (size_t)gridDim.x * 128;
  for (size_t i = (size_t)blockIdx.x * 128 + ch; i < (size_t)NPTS * FOUT; i += stride) {
    const float v = fmaxf(0.f, (y[i] - m) * r * g + be);
    y2[i] = v; s += v; q += v * v;
  }
  atomicAdd(stats + 5 * 512 + ch, s);
  atomicAdd(stats + 5 * 512 + 128 + ch, q);
}

// ---------------- xs0[b][f] = max_n x[b][f][n] ----------------
__global__ void xmax_kernel(const float* __restrict__ x, float* __restrict__ xs0) {
  __shared__ float red[256];
  const float* row = x + (size_t)blockIdx.x * N_;
  float m = -3.4e38f;
  for (int n = threadIdx.x; n < N_; n += 256) m = fmaxf(m, row[n]);
  red[threadIdx.x] = m;
  __syncthreads();
  for (int s = 128; s > 0; s >>= 1) {
    if (threadIdx.x < s) red[threadIdx.x] = fmaxf(red[threadIdx.x], red[threadIdx.x + s]);
    __syncthreads();
  }
  if (threadIdx.x == 0) xs0[blockIdx.x] = red[0];
}

// ---------------- tiny FC branch (batch of 8) in one block ----------------
__global__ __launch_bounds__(256)
void fc_kernel(const float* __restrict__ xs0,
               const float* __restrict__ w1, const float* __restrict__ b1,
               const float* __restrict__ g1, const float* __restrict__ be1,
               const float* __restrict__ w2, const float* __restrict__ b2,
               const float* __restrict__ g2, const float* __restrict__ be2,
               float* __restrict__ xs2) {
  __shared__ float z1[8][64];
  __shared__ float h1[8][64];
  __shared__ float z2[8][128];
  const int t = threadIdx.x;
  for (int i = t; i < 8 * 64; i += 256) {
    const int b = i >> 6, o = i & 63;
    float a = b1[o];
    for (int c = 0; c < 64; ++c) a += xs0[b * 64 + c] * w1[o * 64 + c];
    z1[b][o] = a;
  }
  __syncthreads();
  if (t < 64) {
    float s = 0.f, q = 0.f;
    for (int b = 0; b < 8; ++b) { const float v = z1[b][t]; s += v; q += v * v; }
    const float m = s / 8.f, var = q / 8.f - m * m, r = rsqrtf(var + 1e-5f);
    for (int b = 0; b < 8; ++b) h1[b][t] = lrelu_f((z1[b][t] - m) * r * g1[t] + be1[t]);
  }
  __syncthreads();
  for (int i = t; i < 8 * 128; i += 256) {
    const int b = i >> 7, o = i & 127;
    float a = b2[o];
    for (int c = 0; c < 64; ++c) a += h1[b][c] * w2[o * 64 + c];
    z2[b][o] = a;
  }
  __syncthreads();
  if (t < 128) {
    float s = 0.f, q = 0.f;
    for (int b = 0; b < 8; ++b) { const float v = z2[b][t]; s += v; q += v * v; }
    const float m = s / 8.f, var = q / 8.f - m * m, r = rsqrtf(var + 1e-5f);
    for (int b = 0; b < 8; ++b) xs2[b * 128 + t] = lrelu_f((z2[b][t] - m) * r * g2[t] + be2[t]);
  }
}

// ---------------- ac GEMM: [xs2 | lrelu(bn_uc(y2))] (256) -> 128, WMMA ----------------
__global__ __launch_bounds__(32)
void ac_kernel(const float* __restrict__ y2, const float* __restrict__ xs2,
               const float* __restrict__ ucg, const float* __restrict__ ucbe,
               const float* __restrict__ acw, const float* __restrict__ acb,
               float* __restrict__ stats, float* __restrict__ zac) {
  __shared__ float A[16][260];
  const int lane = threadIdx.x, n16 = lane & 15, kh = lane >> 4;
  float say[8], qay[8];
#pragma unroll
  for (int nt = 0; nt < 8; ++nt) { say[nt] = 0.f; qay[nt] = 0.f; }
  for (int ti = 0; ti < 8; ++ti) {
    const int p0 = (blockIdx.x * 8 + ti) * 16;
    __syncthreads();
    for (int t = lane; t < 16 * 256; t += 32) {
      const int m = t >> 8, c = t & 255;
      const int p = p0 + m, b = p >> 12;
      float v;
      if (c < 128) v = xs2[b * 128 + c];
      else {
        const int cc = c - 128;
        const float z = y2[(size_t)p * FOUT + cc];
        v = lrelu_f((z - stats[5 * 512 + 256 + cc]) * stats[5 * 512 + 384 + cc] * ucg[cc] + ucbe[cc]);
      }
      A[m][c] = v;
    }
    __syncthreads();
    for (int nt = 0; nt < 8; ++nt) {
      v8f acc = {};
      const int o = nt * 16 + n16;
      for (int k0 = 0; k0 < 256; k0 += 4) {
        v2f a, bb;
        a.x  = A[n16][k0 + 2 * kh];           a.y  = A[n16][k0 + 2 * kh + 1];
        bb.x = acw[o * 256 + k0 + 2 * kh];    bb.y = acw[o * 256 + k0 + 2 * kh + 1];
        acc = wmma4(a, bb, acc);
      }
#pragma unroll
      for (int r = 0; r < 8; ++r) {
        const int p = p0 + r + 8 * kh;
        const float z = acc[r] + acb[o];
        zac[(size_t)p * FOUT + o] = z;
        say[nt] += z; qay[nt] += z * z;
      }
    }
  }
  for (int nt = 0; nt < 8; ++nt) {
    const int o = nt * 16 + n16;
    atomicAdd(stats + 8 * 512 + o, say[nt]);
    atomicAdd(stats + 8 * 512 + 128 + o, qay[nt]);
  }
}

// ---------------- final BN + lrelu + transpose to (B, Fout, N) ----------------
__global__ void out_kernel(const float* __restrict__ zac, const float* __restrict__ stats,
                           const float* __restrict__ acg, const float* __restrict__ acbe,
                           float* __restrict__ out) {
  const size_t i = (size_t)blockIdx.x * 256 + threadIdx.x;
  if (i >= (size_t)B_ * FOUT * N_) return;
  const int n = (int)(i & (N_ - 1));
  const size_t bo = i >> 12;
  const int o = (int)(bo & (FOUT - 1));
  const int b = (int)(bo >> 7);
  const float z = zac[(((size_t)b << 12) + n) * FOUT + o];
  out[i] = lrelu_f((z - stats[8 * 512 + 256 + o]) * stats[8 * 512 + 384 + o] * acg[o] + acbe[o]);
}

// ---------------- host launch ----------------
extern "C" void kernel_launch(void* const* d_in, const int* in_sizes, int n_in,
                              void* d_out, int out_size, void* d_ws, size_t ws_size,
                              hipStream_t stream) {
  (void)in_sizes; (void)n_in; (void)out_size; (void)ws_size;
  const float* x      = (const float*)d_in[0];
  const float* cf_w1  = (const float*)d_in[1];
  const float* cf_b1  = (const float*)d_in[2];
  const float* cf_g1  = (const float*)d_in[3];
  const float* cf_be1 = (const float*)d_in[4];
  const float* cf_w2  = (const float*)d_in[5];
  const float* cf_b2  = (const float*)d_in[6];
  const float* cf_g2  = (const float*)d_in[7];
  const float* cf_be2 = (const float*)d_in[8];
  const float* cf_w3  = (const float*)d_in[9];
  const float* cf_b3  = (const float*)d_in[10];
  const float* cf_g3  = (const float*)d_in[11];
  const float* cf_be3 = (const float*)d_in[12];
  const float* ic_w   = (const float*)d_in[13];
  const float* ic_b   = (const float*)d_in[14];
  const float* ic_g   = (const float*)d_in[15];
  const float* ic_be  = (const float*)d_in[16];
  const float* c2_w   = (const float*)d_in[17];
  const float* c2_b   = (const float*)d_in[18];
  const float* c2_g   = (const float*)d_in[19];
  const float* c2_be  = (const float*)d_in[20];
  const float* uc_g   = (const float*)d_in[21];
  const float* uc_be  = (const float*)d_in[22];
  const float* fc1_w  = (const float*)d_in[23];
  const float* fc1_b  = (const float*)d_in[24];
  const float* fc1_g  = (const float*)d_in[25];
  const float* fc1_be = (const float*)d_in[26];
  const float* fc2_w  = (const float*)d_in[27];
  const float* fc2_b  = (const float*)d_in[28];
  const float* fc2_g  = (const float*)d_in[29];
  const float* fc2_be = (const float*)d_in[30];
  const float* ac_w   = (const float*)d_in[31];
  const float* ac_b   = (const float*)d_in[32];
  const float* ac_g   = (const float*)d_in[33];
  const float* ac_be  = (const float*)d_in[34];

  float* ws    = (float*)d_ws;
  float* stats = ws;                                   // 4608 f
  float* xt    = ws + 4608;                            // 2,097,152 f
  float* sq    = ws + 2101760;                         // 32,768 f
  int*   nbr   = (int*)(ws + 2134528);                 // 655,360 i32
  float* y     = ws + 2789888;                         // 4,194,304 f
  float* y2    = ws + 6984192;                         // 4,194,304 f
  float* zac   = ws + 11178496;                        // 4,194,304 f
  float* xs0   = ws + 15372800;                        // 512 f
  float* xs2   = ws + 15373312;                        // 1024 f

  zero_kernel<<<(STATS_F + 255) / 256, 256, 0, stream>>>(stats, STATS_F);
  prep_kernel<<<NPTS / 256, 256, 0, stream>>>(x, xt, sq);
  knn_kernel<<<B_ * (N_ / 16), 32, 0, stream>>>(x, xt, sq, nbr);

#define EDGE_ARGS xt, nbr, cf_w1, cf_b1, cf_g1, cf_be1, cf_w2, cf_b2, cf_g2, cf_be2, \
                  cf_w3, cf_b3, cf_g3, cf_be3, ic_w, ic_b, ic_g, ic_be, c2_w, c2_b, stats, y

  edge_stage_kernel<1><<<NPTS / 8, 32, 0, stream>>>(EDGE_ARGS);
  finalize_kernel<<<1, 128, 0, stream>>>(stats, 0, 16, (float)NEDGE);
  edge_stage_kernel<2><<<NPTS / 8, 32, 0, stream>>>(EDGE_ARGS);
  finalize_kernel<<<1, 128, 0, stream>>>(stats, 1, 64, (float)NEDGE);
  edge_stage_kernel<3><<<NPTS / 8, 32, 0, stream>>>(EDGE_ARGS);
  finalize_kernel<<<1, 128, 0, stream>>>(stats, 2, 64, (float)NEDGE);
  finalize_kernel<<<1, 128, 0, stream>>>(stats, 3, 64, (float)NEDGE);
  edge_stage_kernel<4><<<NPTS / 8, 32, 0, stream>>>(EDGE_ARGS);
  finalize_kernel<<<1, 128, 0, stream>>>(stats, 4, 128, (float)NPTS);

  y2_kernel<<<2048, 128, 0, stream>>>(y, c2_g, c2_be, stats, y2);
  finalize_kernel<<<1, 128, 0, stream>>>(stats, 5, 128, (float)NPTS);

  xmax_kernel<<<B_ * FIN, 256, 0, stream>>>(x, xs0);
  fc_kernel<<<1, 256, 0, stream>>>(xs0, fc1_w, fc1_b, fc1_g, fc1_be,
                                   fc2_w, fc2_b, fc2_g, fc2_be, xs2);

  ac_kernel<<<NPTS / 128, 32, 0, stream>>>(y2, xs2, uc_g, uc_be, ac_w, ac_b, stats, zac);
  finalize_kernel<<<1, 128, 0, stream>>>(stats, 8, 128, (float)NPTS);

  out_kernel<<<(B_ * FOUT * N_ + 255) / 256, 256, 0, stream>>>(zac, stats, ac_g, ac_be,
                                                               (float*)d_out);
#undef EDGE_ARGS
}